// GCN_18021682774333
// MI455X (gfx1250) — compile-verified
//
#include <hip/hip_runtime.h>
#include <hip/hip_bf16.h>

typedef __attribute__((ext_vector_type(2))) float v2f;
typedef __attribute__((ext_vector_type(8))) float v8f;

#define NN 200000      // nodes
#define NE 1250000     // edges
#define NG 2048        // graphs
#define EMB 32

// ---------------- degree / normalization ----------------
__global__ void deg_init_k(float* deg) {
    int i = blockIdx.x * blockDim.x + threadIdx.x;
    if (i < NN) deg[i] = 1.0f;   // self-loop
}
__global__ void deg_edge_k(const int* __restrict__ dst, float* deg) {
    int e = blockIdx.x * blockDim.x + threadIdx.x;
    if (e < NE) atomicAdd(&deg[dst[e]], 1.0f);
}
__global__ void deg_rsqrt_k(float* deg) {
    int i = blockIdx.x * blockDim.x + threadIdx.x;
    if (i < NN) deg[i] = rsqrtf(deg[i]);   // deg >= 1 always
}

// ---------------- fp32 WMMA GEMM: out[M,N] = (relu?)A[M,K] @ W[K,N] --------
// One wave computes a 16-row tile across all N (N/16 accumulator tiles).
// A-frag (16x4 f32): lane L<16 holds A[M=L][k0+0..1] in v0..v1,
//                    lane L>=16 holds A[M=L-16][k0+2..3]   (ISA 7.12.2)
// B-frag (4x16):     mirrored: lanes 0-15 rows k0,k0+1; lanes 16-31 rows k0+2,k0+3
// C/D (16x16 f32):   VGPR j -> M=j (lanes 0-15) / M=j+8 (lanes 16-31), N=lane&15
template<int K, int N, bool RELU_IN>
__global__ void gemm_wmma_k(const float* __restrict__ A, const float* __restrict__ W,
                            float* __restrict__ out, int mtiles) {
    constexpr int NT = N / 16;
    int wave = (int)((blockIdx.x * blockDim.x + threadIdx.x) >> 5);
    int lane = threadIdx.x & 31;
    if (wave >= mtiles) return;               // uniform per wave -> EXEC stays all-1s

    const int m  = lane & 15;
    const int kh = lane >> 4;                 // which K-half of the fragment
    const long long row = (long long)wave * 16 + m;
    const float* __restrict__ arow = A + row * K;

    v8f acc[NT];
#pragma unroll
    for (int nt = 0; nt < NT; ++nt) acc[nt] = (v8f){0.f,0.f,0.f,0.f,0.f,0.f,0.f,0.f};

#pragma unroll 4
    for (int k0 = 0; k0 < K; k0 += 4) {
        float2 av = *reinterpret_cast<const float2*>(arow + k0 + kh * 2);
        if (RELU_IN) { av.x = fmaxf(av.x, 0.f); av.y = fmaxf(av.y, 0.f); }
        v2f a = {av.x, av.y};
#pragma unroll
        for (int nt = 0; nt < NT; ++nt) {
            v2f b = { W[(k0 + kh * 2    ) * N + nt * 16 + m],
                      W[(k0 + kh * 2 + 1) * N + nt * 16 + m] };
            acc[nt] = __builtin_amdgcn_wmma_f32_16x16x4_f32(
                false, a, false, b, (short)0, acc[nt], false, false);
        }
    }

    const long long rbase = (long long)wave * 16 + kh * 8;
#pragma unroll
    for (int nt = 0; nt < NT; ++nt) {
        union { v8f v; float f[8]; } u; u.v = acc[nt];
#pragma unroll
        for (int j = 0; j < 8; ++j)
            out[(rbase + j) * N + nt * 16 + m] = u.f[j];
    }
}

// ---------------- GCN aggregation ----------------
// out[i,c] = dis[i]^2 * t[i,c] + b[c]   (self-loop + bias), then edge atomics add
template<int C>
__global__ void agg_init_k(const float* __restrict__ t, const float* __restrict__ dis,
                           const float* __restrict__ bias, float* __restrict__ out) {
    long long idx = (long long)blockIdx.x * blockDim.x + threadIdx.x;
    if (idx >= (long long)NN * C) return;
    int i = (int)(idx / C);
    int c = (int)(idx % C);
    float d = dis[i];
    out[idx] = d * d * t[idx] + bias[c];
}

// out[dst,c] += dis[src]*dis[dst]*t[src,c]; channel-major -> coalesced lanes
template<int C>
__global__ void agg_edge_k(const float* __restrict__ t, const float* __restrict__ dis,
                           const int* __restrict__ src, const int* __restrict__ dst,
                           float* __restrict__ out) {
    long long idx = (long long)blockIdx.x * blockDim.x + threadIdx.x;
    if (idx >= (long long)NE * C) return;
    int e = (int)(idx / C);
    int c = (int)(idx % C);
    int s = src[e], d = dst[e];
    float nrm = dis[s] * dis[d];
    atomicAdd(&out[(long long)d * C + c], nrm * t[(long long)s * C + c]);
}

// ---------------- mean pool + classifier ----------------
__global__ void pool_zero_k(float* sums, float* cnt) {
    int i = blockIdx.x * blockDim.x + threadIdx.x;
    if (i < NG * EMB) sums[i] = 0.f;
    if (i < NG) cnt[i] = 0.f;
}
__global__ void pool_accum_k(const float* __restrict__ h, const int* __restrict__ batch,
                             float* sums) {
    long long idx = (long long)blockIdx.x * blockDim.x + threadIdx.x;
    if (idx >= (long long)NN * EMB) return;
    int i = (int)(idx >> 5);
    int c = (int)(idx & 31);
    int g = batch[i];
    atomicAdd(&sums[g * EMB + c], fmaxf(h[idx], 0.f));   // relu fused before pooling
}
__global__ void pool_count_k(const int* __restrict__ batch, float* cnt) {
    int i = blockIdx.x * blockDim.x + threadIdx.x;
    if (i < NN) atomicAdd(&cnt[batch[i]], 1.0f);
}
__global__ void classifier_k(const float* __restrict__ sums, const float* __restrict__ cnt,
                             const float* __restrict__ Wc1, const float* __restrict__ bc1,
                             const float* __restrict__ Wc2, const float* __restrict__ bc2,
                             float* __restrict__ outp) {
    int g = blockIdx.x * blockDim.x + threadIdx.x;
    if (g >= NG) return;
    float inv = 1.0f / fmaxf(cnt[g], 1.0f);
    float emb[EMB];
#pragma unroll
    for (int c = 0; c < EMB; ++c) emb[c] = sums[g * EMB + c] * inv;
    float o = bc2[0];
#pragma unroll
    for (int j = 0; j < 16; ++j) {
        float z = bc1[j];
#pragma unroll
        for (int c = 0; c < EMB; ++c) z += emb[c] * Wc1[c * 16 + j];
        o += fmaxf(z, 0.f) * Wc2[j];
    }
    outp[g] = o;
}

// ---------------- launch ----------------
extern "C" void kernel_launch(void* const* d_in, const int* in_sizes, int n_in,
                              void* d_out, int out_size, void* d_ws, size_t ws_size,
                              hipStream_t stream) {
    const float* x   = (const float*)d_in[0];
    const int*   ei  = (const int*)d_in[1];
    const int*   src = ei;
    const int*   dst = ei + NE;
    const int*   bat = (const int*)d_in[2];
    const float* W1  = (const float*)d_in[3];   const float* b1  = (const float*)d_in[4];
    const float* W2  = (const float*)d_in[5];   const float* b2  = (const float*)d_in[6];
    const float* W3  = (const float*)d_in[7];   const float* b3  = (const float*)d_in[8];
    const float* Wc1 = (const float*)d_in[9];   const float* bc1 = (const float*)d_in[10];
    const float* Wc2 = (const float*)d_in[11];  const float* bc2 = (const float*)d_in[12];
    float* out = (float*)d_out;

    // workspace carve-up (~104 MB)
    float* bufA = (float*)d_ws;                       // [NN,64]
    float* bufB = bufA + (size_t)NN * 64;             // [NN,64]
    float* deg  = bufB + (size_t)NN * 64;             // [NN]  (becomes deg^-1/2)
    float* sums = deg + NN;                           // [NG,EMB]
    float* cnt  = sums + (size_t)NG * EMB;            // [NG]

    const int T = 256;
    const int mtiles = NN / 16;                       // 12500 exactly
    const int gemm_blocks = (mtiles + 7) / 8;         // 8 waves / block

    // degree + normalization
    deg_init_k <<<(NN + T - 1) / T, T, 0, stream>>>(deg);
    deg_edge_k <<<(NE + T - 1) / T, T, 0, stream>>>(dst, deg);
    deg_rsqrt_k<<<(NN + T - 1) / T, T, 0, stream>>>(deg);

    // layer 1: x[NN,128] @ W1[128,64]
    gemm_wmma_k<128, 64, false><<<gemm_blocks, T, 0, stream>>>(x, W1, bufA, mtiles);
    agg_init_k<64><<<(int)(((long long)NN * 64 + T - 1) / T), T, 0, stream>>>(bufA, deg, b1, bufB);
    agg_edge_k<64><<<(int)(((long long)NE * 64 + T - 1) / T), T, 0, stream>>>(bufA, deg, src, dst, bufB);

    // layer 2: relu(h1)[NN,64] @ W2[64,64]
    gemm_wmma_k<64, 64, true><<<gemm_blocks, T, 0, stream>>>(bufB, W2, bufA, mtiles);
    agg_init_k<64><<<(int)(((long long)NN * 64 + T - 1) / T), T, 0, stream>>>(bufA, deg, b2, bufB);
    agg_edge_k<64><<<(int)(((long long)NE * 64 + T - 1) / T), T, 0, stream>>>(bufA, deg, src, dst, bufB);

    // layer 3: relu(h2)[NN,64] @ W3[64,32]
    gemm_wmma_k<64, 32, true><<<gemm_blocks, T, 0, stream>>>(bufB, W3, bufA, mtiles);
    agg_init_k<32><<<(int)(((long long)NN * 32 + T - 1) / T), T, 0, stream>>>(bufA, deg, b3, bufB);
    agg_edge_k<32><<<(int)(((long long)NE * 32 + T - 1) / T), T, 0, stream>>>(bufA, deg, src, dst, bufB);

    // mean pool (relu fused) + classifier
    pool_zero_k <<<(NG * EMB + T - 1) / T, T, 0, stream>>>(sums, cnt);
    pool_accum_k<<<(int)(((long long)NN * EMB + T - 1) / T), T, 0, stream>>>(bufB, bat, sums);
    pool_count_k<<<(NN + T - 1) / T, T, 0, stream>>>(bat, cnt);
    classifier_k<<<(NG + T - 1) / T, T, 0, stream>>>(sums, cnt, Wc1, bc1, Wc2, bc2, out);
}